// GeometricAttention_86431921865223
// MI455X (gfx1250) — compile-verified
//
#include <hip/hip_runtime.h>
#include <math.h>

typedef float v2f __attribute__((ext_vector_type(2)));
typedef float v8f __attribute__((ext_vector_type(8)));

#define Bn   4
#define Nn   4096
#define Dn   256
#define Hn   8
#define DHn  32
#define Kn   16

// ---------------------------------------------------------------------------
// GEMM: C[M,N] = A[M,K] @ W[K,N] (+ bias), f32 WMMA 16x16x4.  K is fixed 256.
// Block = 256 threads (8 waves) computes a 256x64 tile of C.
// The K x 64 weight panel (64KB) is staged once per block into LDS with
// GLOBAL_LOAD_ASYNC_TO_LDS_B128 (ASYNCcnt), then shared by all 8 waves.
// Each wave owns 32 rows: 2 A fragments x 4 N-tiles = 8 f32 accumulators;
// every B fragment read from LDS feeds two WMMAs.
// Fragment layouts per ISA (32-bit 16x4 A): lanes 0-15 = M rows with K{0,1}
// in the two VGPRs, lanes 16-31 hold K{2,3}. B mirrored (lanes = N columns).
// C/D: 8 VGPRs, lane half selects M rows 0-7 / 8-15.
// ---------------------------------------------------------------------------
__global__ void ga_wmma_gemm_bias(const float* __restrict__ A,
                                  const float* __restrict__ W,
                                  const float* __restrict__ bias,
                                  float* __restrict__ C,
                                  int M, int Nd) {
  __shared__ float sW[Dn * 64];               // [k][64] panel, 64KB

  const int tid  = threadIdx.x;
  const int n0   = blockIdx.y * 64;

  // ---- async stage of W panel: 4096 float4 chunks, 16 per thread ----------
  for (int p = tid; p < Dn * 16; p += 256) {
    const int k = p >> 4, c = p & 15;
    unsigned long long srcv =
        (unsigned long long)(W + (size_t)k * Nd + n0 + (c << 2));
    unsigned dstv = (unsigned)(unsigned long long)(sW + (p << 2)); // LDS byte off
    asm volatile("global_load_async_to_lds_b128 %0, %1, off"
                 :: "v"(dstv), "v"(srcv) : "memory");
  }
  asm volatile("s_wait_asynccnt 0x0" ::: "memory");
  __syncthreads();

  // ---- compute ------------------------------------------------------------
  const int wv   = tid >> 5;
  const int lane = tid & 31;
  const int m0   = blockIdx.x * 256 + wv * 32;

  v8f acc[2][4] = {};

  const int ak = (lane >> 4) * 2;             // K offset for this lane half
  const float* __restrict__ Arow0 = A + (size_t)(m0 + (lane & 15)) * Dn + ak;
  const float* __restrict__ Arow1 = Arow0 + (size_t)16 * Dn;
  const int ln = lane & 15;                   // local N within a 16-tile
  const int bk = (lane >> 4) * 2;

  for (int kb = 0; kb < Dn; kb += 4) {
    __builtin_prefetch(Arow0 + kb + 16);      // global_prefetch_b8 (A stream)
    v2f a0 = *(const v2f*)(Arow0 + kb);
    v2f a1 = *(const v2f*)(Arow1 + kb);
#pragma unroll
    for (int t = 0; t < 4; ++t) {
      v2f b;
      b.x = sW[(kb + bk)     * 64 + t * 16 + ln];   // W[kb+bk  ][n]
      b.y = sW[(kb + bk + 1) * 64 + t * 16 + ln];   // W[kb+bk+1][n]
      acc[0][t] = __builtin_amdgcn_wmma_f32_16x16x4_f32(
          false, a0, false, b, (short)0, acc[0][t], false, false);
      acc[1][t] = __builtin_amdgcn_wmma_f32_16x16x4_f32(
          false, a1, false, b, (short)0, acc[1][t], false, false);
    }
  }

  // ---- epilogue -----------------------------------------------------------
#pragma unroll
  for (int mt = 0; mt < 2; ++mt) {
    const int mBase = m0 + mt * 16 + ((lane >> 4) ? 8 : 0);
#pragma unroll
    for (int t = 0; t < 4; ++t) {
      const int cn = n0 + t * 16 + ln;
      const float bv = bias ? bias[cn] : 0.0f;
#pragma unroll
      for (int r = 0; r < 8; ++r)
        C[(size_t)(mBase + r) * Nd + cn] = acc[mt][t][r] + bv;
    }
  }
}

// ---------------------------------------------------------------------------
// KNN: brute-force top-16 smallest d2 (stable ties = lower index).
// One workgroup (256 threads) per 256 query points; whole batch's points +
// squared norms staged in 64KB LDS (WGP has 320KB).
// ---------------------------------------------------------------------------
__global__ void ga_knn_kernel(const float* __restrict__ points,
                              int* __restrict__ knn) {
  __shared__ float sx[Nn], sy[Nn], sz[Nn], ssq[Nn];
  const int tid = threadIdx.x;
  const int b = blockIdx.y;
  const float* pb = points + (size_t)b * Nn * 3;

  for (int t = tid; t < Nn; t += 256) {
    float x = pb[3 * t + 0], y = pb[3 * t + 1], z = pb[3 * t + 2];
    sx[t] = x; sy[t] = y; sz[t] = z;
    ssq[t] = x * x + y * y + z * z;
  }
  __syncthreads();

  const int iq = blockIdx.x * 256 + tid;
  const float qx = sx[iq], qy = sy[iq], qz = sz[iq], sqi = ssq[iq];

  float best[Kn];
  int bidx[Kn];
#pragma unroll
  for (int s = 0; s < Kn; ++s) { best[s] = 3.0e38f; bidx[s] = 0; }

  for (int j = 0; j < Nn; ++j) {
    float d2 = sqi + ssq[j] - 2.0f * (qx * sx[j] + qy * sy[j] + qz * sz[j]);
    if (d2 < best[Kn - 1]) {
      float cd = d2; int cj = j;
#pragma unroll
      for (int s = 0; s < Kn; ++s) {          // register-only stable insert
        if (cd < best[s]) {
          float tf = best[s]; best[s] = cd; cd = tf;
          int   ti = bidx[s]; bidx[s] = cj; cj = ti;
        }
      }
    }
  }

  int* out = knn + ((size_t)b * Nn + iq) * Kn;
#pragma unroll
  for (int s = 0; s < Kn; ++s) out[s] = bidx[s];
}

// ---------------------------------------------------------------------------
// Attention: one wave32 per (b,h,i); lane = channel d (dh=32).
// pos_enc / geo_w / normal_sim recomputed inline (saves ~35MB HBM traffic).
// ---------------------------------------------------------------------------
__device__ __forceinline__ float ga_wave_sum(float v) {
#pragma unroll
  for (int off = 16; off > 0; off >>= 1) v += __shfl_xor(v, off, 32);
  return v;
}

__global__ void ga_attn_kernel(const float* __restrict__ qkv,
                               const int* __restrict__ knn,
                               const float* __restrict__ points,
                               const float* __restrict__ geo,
                               const float* __restrict__ w_geo,
                               const float* __restrict__ b_geo,
                               const float* __restrict__ w_pos,
                               const float* __restrict__ b_pos,
                               float* __restrict__ out_heads) {
  const int wave = blockIdx.x * (blockDim.x >> 5) + (threadIdx.x >> 5);
  const int lane = threadIdx.x & 31;
  const int i = wave & (Nn - 1);
  const int h = (wave >> 12) & (Hn - 1);
  const int b = wave >> 15;

  const size_t row = (size_t)b * Nn + i;
  const float q = qkv[row * (3 * Dn) + h * DHn + lane];

  // lane-resident pos-MLP weights
  const float wpa = w_pos[0 * DHn + lane];
  const float wpb = w_pos[1 * DHn + lane];
  const float wpc = w_pos[2 * DHn + lane];
  const float bp  = b_pos[lane];

  const float* gf = geo + row * 4;
  const float g0 = gf[0], g1 = gf[1], g2 = gf[2], g3 = gf[3];
  const float gw = g0 * w_geo[0 * Hn + h] + g1 * w_geo[1 * Hn + h] +
                   g2 * w_geo[2 * Hn + h] + g3 * w_geo[3 * Hn + h] + b_geo[h];
  const float ni = fmaxf(sqrtf(g0 * g0 + g1 * g1 + g2 * g2), 1e-8f);

  const float pix = points[row * 3 + 0];
  const float piy = points[row * 3 + 1];
  const float piz = points[row * 3 + 2];

  const int* kn = knn + row * Kn;
  const float scale = 0.0625f;                // 256^-0.5

  float dots[Kn], vv[Kn], pe[Kn];
#pragma unroll
  for (int kk = 0; kk < Kn; ++kk) {
    const int j = kn[kk];
    const size_t jr = (size_t)b * Nn + j;
    const float* krow = qkv + jr * (3 * Dn);
    const float kv = krow[Dn + h * DHn + lane];
    vv[kk] = krow[2 * Dn + h * DHn + lane];

    const float rx = pix - points[jr * 3 + 0];
    const float ry = piy - points[jr * 3 + 1];
    const float rz = piz - points[jr * 3 + 2];
    const float p = fmaxf(rx * wpa + ry * wpb + rz * wpc + bp, 0.0f);
    pe[kk] = p;

    // fused dual reduction: sum_d q*(k*scale + 0.5*pos)
    const float d = ga_wave_sum(q * (kv * scale + 0.5f * p));

    const float* gfj = geo + jr * 4;
    const float num = g0 * gfj[0] + g1 * gfj[1] + g2 * gfj[2];
    const float nj = fmaxf(
        sqrtf(gfj[0] * gfj[0] + gfj[1] * gfj[1] + gfj[2] * gfj[2]), 1e-8f);
    dots[kk] = d + 0.3f * (num / (ni * nj)) * gw;
  }

  float m = dots[0];
#pragma unroll
  for (int kk = 1; kk < Kn; ++kk) m = fmaxf(m, dots[kk]);
  float se = 0.0f;
#pragma unroll
  for (int kk = 0; kk < Kn; ++kk) { dots[kk] = expf(dots[kk] - m); se += dots[kk]; }
  const float inv = 1.0f / se;

  float o = 0.0f;
#pragma unroll
  for (int kk = 0; kk < Kn; ++kk) o += dots[kk] * (vv[kk] + pe[kk]);

  out_heads[row * Dn + h * DHn + lane] = o * inv;
}

// ---------------------------------------------------------------------------
// Host orchestration. Inputs (setup_inputs order, all f32):
// 0:x 1:points 2:geo_features 3:w_qkv 4:w_geo 5:b_geo 6:w_pos 7:b_pos
// 8:w_proj 9:b_proj
// ---------------------------------------------------------------------------
extern "C" void kernel_launch(void* const* d_in, const int* in_sizes, int n_in,
                              void* d_out, int out_size, void* d_ws, size_t ws_size,
                              hipStream_t stream) {
  const float* x      = (const float*)d_in[0];
  const float* points = (const float*)d_in[1];
  const float* geo    = (const float*)d_in[2];
  const float* w_qkv  = (const float*)d_in[3];
  const float* w_geo  = (const float*)d_in[4];
  const float* b_geo  = (const float*)d_in[5];
  const float* w_pos  = (const float*)d_in[6];
  const float* b_pos  = (const float*)d_in[7];
  const float* w_proj = (const float*)d_in[8];
  const float* b_proj = (const float*)d_in[9];
  float* out = (float*)d_out;

  const int M = Bn * Nn;                      // 16384 rows
  float* qkv       = (float*)d_ws;            // M * 768
  int*   knn       = (int*)(qkv + (size_t)M * 3 * Dn);   // M * 16
  float* out_heads = (float*)(knn + (size_t)M * Kn);     // M * 256

  // 1) qkv = x @ w_qkv  (16384 x 256 x 768), no bias
  ga_wmma_gemm_bias<<<dim3(M / 256, (3 * Dn) / 64), 256, 0, stream>>>(
      x, w_qkv, nullptr, qkv, M, 3 * Dn);
  // 2) KNN top-16
  ga_knn_kernel<<<dim3(Nn / 256, Bn), 256, 0, stream>>>(points, knn);
  // 3) attention: B*H*N waves, 8 waves/block
  ga_attn_kernel<<<(Bn * Hn * Nn) / 8, 256, 0, stream>>>(
      qkv, knn, points, geo, w_geo, b_geo, w_pos, b_pos, out_heads);
  // 4) out = out_heads @ w_proj + b_proj  (16384 x 256 x 256)
  ga_wmma_gemm_bias<<<dim3(M / 256, Dn / 64), 256, 0, stream>>>(
      out_heads, w_proj, b_proj, out, M, Dn);
}